// TimeGNN_18811956756540
// MI455X (gfx1250) — compile-verified
//
#include <hip/hip_runtime.h>

// ---------------- shapes ----------------
#define BB 32
#define SS 128
#define DIN 16
#define HH 128
#define H3 384
#define NROWS (BB*SS)        // 4096

typedef float v2f __attribute__((ext_vector_type(2)));
typedef float v8f __attribute__((ext_vector_type(8)));

static __device__ inline v8f wmma4(v2f a, v2f b, v8f c) {
  // D = A(16x4 f32) * B(4x16 f32) + C(16x16 f32); native CDNA5 f32 matrix op
  return __builtin_amdgcn_wmma_f32_16x16x4_f32(false, a, false, b, (short)0, c, false, false);
}

// ---- CDNA5 async global->LDS DMA (ASYNCcnt-tracked), via inline asm ----
// LDS byte offset = low 32 bits of the generic shared-memory pointer
// (flat->LDS aperture uses addr[31:0] directly, ISA 10.2).
static __device__ inline void cp_async_f4(float* ldst, const float* gsrc) {
  const unsigned loff = (unsigned)(unsigned long long)ldst;
  asm volatile("global_load_async_to_lds_b128 %0, %1, off"
               :: "v"(loff), "v"(gsrc) : "memory");
}
static __device__ inline void cp_async_wait() {
  asm volatile("s_wait_asynccnt 0x0" ::: "memory");
}

// ---------------- workspace layout (floats) ----------------
#define OF_T1   0u
#define OF_T2   (OF_T1  + NROWS*HH)
#define OF_CAT  (OF_T2  + NROWS*HH)
#define OF_X    (OF_CAT + NROWS*H3)
#define OF_HS   (OF_X   + NROWS*HH)
#define OF_HR   (OF_HS  + NROWS*HH)
#define OF_ADJ  (OF_HR  + NROWS*HH)
#define OF_DEG  (OF_ADJ + NROWS*SS)
#define OF_AGG  (OF_DEG + NROWS)
#define OF_OUT  (OF_AGG + NROWS*HH)
#define OF_XN2  (OF_OUT + NROWS*HH)
#define OF_SUM  (OF_XN2 + NROWS*HH)
#define OF_ACC  (OF_SUM + 2*HH)
// total ~6.3M floats = ~25.2 MB

// =====================================================================
// K1: t1 = data @ w11^T ; t2 = data @ w21^T ; cat[:,256:] = data @ w31^T
// =====================================================================
__global__ __launch_bounds__(128) void k_embed(
    const float* __restrict__ data,
    const float* __restrict__ w11, const float* __restrict__ b11,
    const float* __restrict__ w21, const float* __restrict__ b21,
    const float* __restrict__ w31, const float* __restrict__ b31,
    float* __restrict__ t1, float* __restrict__ t2, float* __restrict__ cat)
{
  __shared__ float lds[16*DIN];
  const int m0  = blockIdx.x * 16;
  const int tid = threadIdx.x;
  for (int e4 = tid; e4 < 64; e4 += 128) {            // 16x16 tile as float4
    const int rr = e4 >> 2, cc = (e4 & 3) << 2;
    cp_async_f4(&lds[rr*DIN + cc], &data[(m0 + rr)*DIN + cc]);
  }
  cp_async_wait();
  __syncthreads();

  const int wave = tid >> 5, lane = tid & 31;
  const int m  = lane & 15, k0 = (lane >> 4) << 1;
  const int nl = m,         kk = k0;
  const int hi8 = (lane >> 4) << 3;
  const float* pa = lds + m*DIN + k0;

  for (int job = wave; job < 24; job += 4) {
    const int which = job >> 3, n0 = (job & 7) << 4;
    const float* w    = (which == 0) ? w11 : (which == 1) ? w21 : w31;
    const float* bias = (which == 0) ? b11 : (which == 1) ? b21 : b31;
    const float* pb = w + (n0 + nl)*DIN + kk;
    v8f c = {};
    #pragma unroll
    for (int kb = 0; kb < DIN; kb += 4) {
      v2f a;  a[0]  = pa[kb];  a[1]  = pa[kb + 1];
      v2f bm; bm[0] = pb[kb];  bm[1] = pb[kb + 1];
      c = wmma4(a, bm, c);
    }
    const float bs = bias[n0 + nl];
    if (which == 2) {
      float* p = cat + (m0 + hi8)*H3 + 256 + n0 + nl;
      #pragma unroll
      for (int r = 0; r < 8; ++r) p[r*H3] = c[r] + bs;
    } else {
      float* p = (which ? t2 : t1) + (m0 + hi8)*HH + n0 + nl;
      #pragma unroll
      for (int r = 0; r < 8; ++r) p[r*HH] = c[r] + bs;
    }
  }
}

// =====================================================================
// K2: dilated convs as shifted HxH GEMMs, halo staged via async DMA
// =====================================================================
__global__ __launch_bounds__(128) void k_conv(
    const float* __restrict__ t1, const float* __restrict__ t2,
    const float* __restrict__ w12, const float* __restrict__ b12,
    const float* __restrict__ w22, const float* __restrict__ b22,
    float* __restrict__ cat)
{
  __shared__ float ldsA[22*HH];   // t1 rows s0-3 .. s0+18
  __shared__ float ldsB[36*HH];   // t2 rows s0-10 .. s0+25
  const int s0 = blockIdx.x << 4, b = blockIdx.y;
  const int tid = threadIdx.x;
  const float4 z4 = {0.f, 0.f, 0.f, 0.f};

  for (int e4 = tid; e4 < 22*32; e4 += 128) {
    const int rr = e4 >> 5, cc = (e4 & 31) << 2, s = s0 - 3 + rr;
    float* lp = &ldsA[rr*HH + cc];
    if (s >= 0 && s < SS) cp_async_f4(lp, &t1[(b*SS + s)*HH + cc]);
    else                  *(float4*)lp = z4;
  }
  for (int e4 = tid; e4 < 36*32; e4 += 128) {
    const int rr = e4 >> 5, cc = (e4 & 31) << 2, s = s0 - 10 + rr;
    float* lp = &ldsB[rr*HH + cc];
    if (s >= 0 && s < SS) cp_async_f4(lp, &t2[(b*SS + s)*HH + cc]);
    else                  *(float4*)lp = z4;
  }
  cp_async_wait();
  __syncthreads();

  const int wave = tid >> 5, lane = tid & 31;
  const int m  = lane & 15, k0 = (lane >> 4) << 1;
  const int nl = m,         kk = k0;
  const int hi8 = (lane >> 4) << 3;

  for (int job = wave; job < 16; job += 4) {
    const int path = job >> 3, n0 = (job & 7) << 4;
    v8f c = {};
    if (path == 0) {
      for (int k = 0; k < 3; ++k) {
        const float* pa = ldsA + (k*3 + m)*HH + k0;          // offsets {-3,0,3}
        const float* pb = w12 + (n0 + nl)*(HH*3) + kk*3 + k;
        #pragma unroll 8
        for (int kb = 0; kb < HH; kb += 4) {
          v2f a;  a[0]  = pa[kb];        a[1]  = pa[kb + 1];
          v2f bm; bm[0] = pb[kb*3];      bm[1] = pb[(kb + 1)*3];
          c = wmma4(a, bm, c);
        }
      }
    } else {
      for (int k = 0; k < 5; ++k) {
        const float* pa = ldsB + (k*5 + m)*HH + k0;          // offsets {-10..10}
        const float* pb = w22 + (n0 + nl)*(HH*5) + kk*5 + k;
        #pragma unroll 8
        for (int kb = 0; kb < HH; kb += 4) {
          v2f a;  a[0]  = pa[kb];        a[1]  = pa[kb + 1];
          v2f bm; bm[0] = pb[kb*5];      bm[1] = pb[(kb + 1)*5];
          c = wmma4(a, bm, c);
        }
      }
    }
    const float bs = (path == 0 ? b12 : b22)[n0 + nl];
    float* p = cat + (b*SS + s0 + hi8)*H3 + (path << 7) + n0 + nl;
    #pragma unroll
    for (int r = 0; r < 8; ++r) p[r*H3] = c[r] + bs;
  }
}

// =====================================================================
// K3: x = relu(cat @ wf + bf)   M=4096 K=384 N=128
// =====================================================================
__global__ __launch_bounds__(128) void k_wf(
    const float* __restrict__ cat, const float* __restrict__ wf,
    const float* __restrict__ bf, float* __restrict__ x)
{
  __shared__ float lds[16*H3];
  const int m0 = blockIdx.x << 4, tid = threadIdx.x;
  for (int e4 = tid; e4 < 16*96; e4 += 128) {
    const int rr = e4 / 96, cc = (e4 - rr*96) << 2;
    cp_async_f4(&lds[rr*H3 + cc], &cat[(m0 + rr)*H3 + cc]);
  }
  cp_async_wait();
  __syncthreads();

  const int wave = tid >> 5, lane = tid & 31;
  const int m = lane & 15, k0 = (lane >> 4) << 1;
  const int nl = m, kk = k0;
  const int hi8 = (lane >> 4) << 3;
  const float* pa = lds + m*H3 + k0;

  for (int nt = wave; nt < 8; nt += 4) {
    const int n0 = nt << 4;
    const float* pb = wf + kk*HH + n0 + nl;
    v8f c = {};
    #pragma unroll 8
    for (int kb = 0; kb < H3; kb += 4) {
      v2f a;  a[0]  = pa[kb];        a[1]  = pa[kb + 1];
      v2f bm; bm[0] = pb[kb*HH];     bm[1] = pb[(kb + 1)*HH];
      c = wmma4(a, bm, c);
    }
    const float bs = bf[n0 + nl];
    float* p = x + (m0 + hi8)*HH + n0 + nl;
    #pragma unroll
    for (int r = 0; r < 8; ++r) p[r*HH] = fmaxf(c[r] + bs, 0.f);
  }
}

// =====================================================================
// K4: hs = x @ w1[:128], hr = x @ w1[128:]
// =====================================================================
__global__ __launch_bounds__(128) void k_proj(
    const float* __restrict__ x, const float* __restrict__ w1,
    float* __restrict__ hs, float* __restrict__ hr)
{
  __shared__ float lds[16*HH];
  const int m0 = blockIdx.x << 4, tid = threadIdx.x;
  for (int e4 = tid; e4 < 512; e4 += 128) {
    const int rr = e4 >> 5, cc = (e4 & 31) << 2;
    cp_async_f4(&lds[rr*HH + cc], &x[(m0 + rr)*HH + cc]);
  }
  cp_async_wait();
  __syncthreads();

  const int wave = tid >> 5, lane = tid & 31;
  const int m = lane & 15, k0 = (lane >> 4) << 1;
  const int nl = m, kk = k0;
  const int hi8 = (lane >> 4) << 3;
  const float* pa = lds + m*HH + k0;

  for (int job = wave; job < 16; job += 4) {
    const int half = job >> 3, n0 = (job & 7) << 4;
    const float* pb = w1 + (half*HH + kk)*HH + n0 + nl;
    v8f c = {};
    #pragma unroll 8
    for (int kb = 0; kb < HH; kb += 4) {
      v2f a;  a[0]  = pa[kb];        a[1]  = pa[kb + 1];
      v2f bm; bm[0] = pb[kb*HH];     bm[1] = pb[(kb + 1)*HH];
      c = wmma4(a, bm, c);
    }
    float* p = (half ? hr : hs) + (m0 + hi8)*HH + n0 + nl;
    #pragma unroll
    for (int r = 0; r < 8; ++r) p[r*HH] = c[r];
  }
}

// =====================================================================
// K5: fused Gumbel edge sampling; the (B,S,S,H) tensor never exists.
// =====================================================================
__global__ __launch_bounds__(128) void k_adj(
    const float* __restrict__ hs, const float* __restrict__ hr,
    const float* __restrict__ b1, const float* __restrict__ w2,
    const float* __restrict__ b2, const float* __restrict__ gu,
    float* __restrict__ adj, float* __restrict__ deg)
{
  __shared__ float shr[HH];
  __shared__ float sd[HH];
  const int i = blockIdx.x, b = blockIdx.y, j = threadIdx.x;
  shr[j] = hr[(b*SS + i)*HH + j] + b1[j];
  sd[j]  = w2[j*2 + 0] - w2[j*2 + 1];
  __syncthreads();

  const float* hsr = hs + (b*SS + j)*HH;
  float acc = 0.f;
  #pragma unroll 4
  for (int h = 0; h < HH; h += 4) {
    const float4 v = *(const float4*)(hsr + h);
    acc += fmaxf(v.x + shr[h  ], 0.f) * sd[h  ];
    acc += fmaxf(v.y + shr[h+1], 0.f) * sd[h+1];
    acc += fmaxf(v.z + shr[h+2], 0.f) * sd[h+2];
    acc += fmaxf(v.w + shr[h+3], 0.f) * sd[h+3];
  }
  const float db = b2[0] - b2[1];
  const int gi = ((b*SS + i)*SS + j)*2;
  const float u0 = fminf(fmaxf(gu[gi    ], 1e-6f), 1.f - 1e-6f);
  const float u1 = fminf(fmaxf(gu[gi + 1], 1e-6f), 1.f - 1e-6f);
  const float g0 = -logf(-logf(u0));
  const float g1 = -logf(-logf(u1));
  const float a  = ((acc + db) >= (g1 - g0) && (j > i)) ? 1.f : 0.f;
  adj[(b*SS + i)*SS + j] = a;
  atomicAdd(&deg[b*SS + j], a);
}

// =====================================================================
// K6a: agg[b,j,:] = (adj^T @ xn)[j,:] / max(deg[b,j],1); adj^T tile in LDS
// =====================================================================
__global__ __launch_bounds__(128) void k_agg(
    const float* __restrict__ adj, const float* __restrict__ deg,
    const float* __restrict__ xn, float* __restrict__ agg)
{
  __shared__ float ladj[SS*16];    // [i][j-local]
  const int j0 = blockIdx.x << 4, b = blockIdx.y;
  const int tid = threadIdx.x;
  for (int e4 = tid; e4 < SS*4; e4 += 128) {
    const int i = e4 >> 2, cc = (e4 & 3) << 2;
    cp_async_f4(&ladj[i*16 + cc], &adj[(b*SS + i)*SS + j0 + cc]);
  }
  cp_async_wait();
  __syncthreads();

  const int wave = tid >> 5, lane = tid & 31;
  const int m = lane & 15, k0 = (lane >> 4) << 1;
  const int nl = m, kk = k0;
  const int hi8 = (lane >> 4) << 3;
  const float* pa = ladj + k0*16 + m;

  for (int nt = wave; nt < 8; nt += 4) {
    const int n0 = nt << 4;
    const float* pb = xn + (b*SS + kk)*HH + n0 + nl;
    v8f c = {};
    #pragma unroll 8
    for (int kb = 0; kb < SS; kb += 4) {
      v2f a;  a[0]  = pa[kb*16];     a[1]  = pa[(kb + 1)*16];
      v2f bm; bm[0] = pb[kb*HH];     bm[1] = pb[(kb + 1)*HH];
      c = wmma4(a, bm, c);
    }
    float* p = agg + (b*SS + j0 + hi8)*HH + n0 + nl;
    const float* dp = deg + b*SS + j0 + hi8;
    #pragma unroll
    for (int r = 0; r < 8; ++r) p[r*HH] = c[r] / fmaxf(dp[r], 1.f);
  }
}

// =====================================================================
// K6b: out = agg @ wl[l] + xn @ wr[l] + bl[l]; BN stats via f32 atomics
// =====================================================================
__global__ __launch_bounds__(128) void k_gcn(
    const float* __restrict__ agg, const float* __restrict__ xn,
    const float* __restrict__ wl, const float* __restrict__ bl,
    const float* __restrict__ wr, float* __restrict__ out,
    float* __restrict__ sums, int l)
{
  __shared__ float la[16*HH];
  __shared__ float lx[16*HH];
  const int m0 = blockIdx.x << 4, tid = threadIdx.x;
  for (int e4 = tid; e4 < 512; e4 += 128) {
    const int rr = e4 >> 5, cc = (e4 & 31) << 2;
    cp_async_f4(&la[rr*HH + cc], &agg[(m0 + rr)*HH + cc]);
    cp_async_f4(&lx[rr*HH + cc], &xn [(m0 + rr)*HH + cc]);
  }
  cp_async_wait();
  __syncthreads();

  const int wave = tid >> 5, lane = tid & 31;
  const int m = lane & 15, k0 = (lane >> 4) << 1;
  const int nl = m, kk = k0;
  const int hi8 = (lane >> 4) << 3;
  const float* Wl = wl + l*HH*HH;
  const float* Wr = wr + l*HH*HH;
  const float* paA = la + m*HH + k0;
  const float* paX = lx + m*HH + k0;

  for (int nt = wave; nt < 8; nt += 4) {
    const int n0 = nt << 4;
    const float* pbL = Wl + kk*HH + n0 + nl;
    const float* pbR = Wr + kk*HH + n0 + nl;
    v8f c = {};
    #pragma unroll 8
    for (int kb = 0; kb < HH; kb += 4) {
      v2f a;  a[0]  = paA[kb];        a[1]  = paA[kb + 1];
      v2f bm; bm[0] = pbL[kb*HH];     bm[1] = pbL[(kb + 1)*HH];
      c = wmma4(a, bm, c);
    }
    #pragma unroll 8
    for (int kb = 0; kb < HH; kb += 4) {
      v2f a;  a[0]  = paX[kb];        a[1]  = paX[kb + 1];
      v2f bm; bm[0] = pbR[kb*HH];     bm[1] = pbR[(kb + 1)*HH];
      c = wmma4(a, bm, c);
    }
    const float bs = bl[l*HH + n0 + nl];
    float* p = out + (m0 + hi8)*HH + n0 + nl;
    float psum = 0.f, psq = 0.f;
    #pragma unroll
    for (int r = 0; r < 8; ++r) {
      const float v = c[r] + bs;
      p[r*HH] = v;
      psum += v; psq += v*v;
    }
    atomicAdd(&sums[n0 + nl], psum);
    atomicAdd(&sums[HH + n0 + nl], psq);
  }
}

// =====================================================================
// K6c: apply BN; accumulate last-seq-position weighted sum for the head
// =====================================================================
__global__ __launch_bounds__(256) void k_bn(
    const float* __restrict__ out, const float* __restrict__ sums,
    const float* __restrict__ gamma, const float* __restrict__ beta,
    const float* __restrict__ wg, float* __restrict__ xnext,
    float* __restrict__ acc, int l)
{
  const int e = blockIdx.x * 256 + threadIdx.x;   // 524288 elems
  const int row = e >> 7, n = e & 127;
  const float inv_n = 1.f / (float)NROWS;
  const float mean = sums[n] * inv_n;
  const float var  = sums[HH + n] * inv_n - mean*mean;
  const float inv  = rsqrtf(var + 1e-5f);
  const float v = (out[e] - mean) * inv * gamma[l*HH + n] + beta[l*HH + n];
  xnext[e] = v;
  if ((row & (SS - 1)) == SS - 1) {               // s == 127
    const int b = row >> 7;
    acc[b*HH + n] += v * wg[l];
  }
}

// =====================================================================
// K7: head: xw=relu(acc+bg); h2=relu(xw@we+be); out=h2@wo+bo
// =====================================================================
__global__ __launch_bounds__(128) void k_head(
    const float* __restrict__ acc, const float* __restrict__ bg,
    const float* __restrict__ we, const float* __restrict__ be,
    const float* __restrict__ wo, const float* __restrict__ bo,
    float* __restrict__ outp)
{
  __shared__ float xw[HH];
  __shared__ float red[64];
  const int b = blockIdx.x, t = threadIdx.x;
  xw[t] = fmaxf(acc[b*HH + t] + bg[0], 0.f);
  __syncthreads();
  if (t < 64) {
    float s = be[t];
    #pragma unroll 4
    for (int k = 0; k < HH; ++k) s += xw[k] * we[k*64 + t];
    red[t] = fmaxf(s, 0.f) * wo[t];
  }
  __syncthreads();
  if (t == 0) {
    float s = bo[0];
    for (int k = 0; k < 64; ++k) s += red[k];
    outp[b] = s;
  }
}

// =====================================================================
extern "C" void kernel_launch(void* const* d_in, const int* in_sizes, int n_in,
                              void* d_out, int out_size, void* d_ws, size_t ws_size,
                              hipStream_t stream) {
  const float* data = (const float*)d_in[0];
  const float* gu   = (const float*)d_in[1];
  const float* w11  = (const float*)d_in[2];
  const float* b11  = (const float*)d_in[3];
  const float* w12  = (const float*)d_in[4];
  const float* b12  = (const float*)d_in[5];
  const float* w21  = (const float*)d_in[6];
  const float* b21  = (const float*)d_in[7];
  const float* w22  = (const float*)d_in[8];
  const float* b22  = (const float*)d_in[9];
  const float* w31  = (const float*)d_in[10];
  const float* b31  = (const float*)d_in[11];
  const float* wf   = (const float*)d_in[12];
  const float* bf   = (const float*)d_in[13];
  const float* w1   = (const float*)d_in[14];
  const float* b1   = (const float*)d_in[15];
  const float* w2   = (const float*)d_in[16];
  const float* b2   = (const float*)d_in[17];
  const float* wl   = (const float*)d_in[18];
  const float* bl   = (const float*)d_in[19];
  const float* wr   = (const float*)d_in[20];
  const float* gmm  = (const float*)d_in[21];
  const float* bta  = (const float*)d_in[22];
  const float* wg   = (const float*)d_in[23];
  const float* bg   = (const float*)d_in[24];
  const float* we   = (const float*)d_in[25];
  const float* be   = (const float*)d_in[26];
  const float* wo   = (const float*)d_in[27];
  const float* bo   = (const float*)d_in[28];

  float* ws   = (float*)d_ws;
  float* t1   = ws + OF_T1;
  float* t2   = ws + OF_T2;
  float* cat  = ws + OF_CAT;
  float* x    = ws + OF_X;
  float* hs   = ws + OF_HS;
  float* hr   = ws + OF_HR;
  float* adj  = ws + OF_ADJ;
  float* deg  = ws + OF_DEG;
  float* agg  = ws + OF_AGG;
  float* outb = ws + OF_OUT;
  float* xn2  = ws + OF_XN2;
  float* sums = ws + OF_SUM;
  float* acc  = ws + OF_ACC;

  hipMemsetAsync(deg, 0, NROWS * sizeof(float), stream);
  hipMemsetAsync(acc, 0, BB * HH * sizeof(float), stream);

  k_embed<<<NROWS/16, 128, 0, stream>>>(data, w11, b11, w21, b21, w31, b31, t1, t2, cat);
  k_conv <<<dim3(SS/16, BB), 128, 0, stream>>>(t1, t2, w12, b12, w22, b22, cat);
  k_wf   <<<NROWS/16, 128, 0, stream>>>(cat, wf, bf, x);
  k_proj <<<NROWS/16, 128, 0, stream>>>(x, w1, hs, hr);
  k_adj  <<<dim3(SS, BB), 128, 0, stream>>>(hs, hr, b1, w2, b2, gu, adj, deg);

  float* cur = x;
  float* nxt = xn2;
  for (int l = 0; l < 3; ++l) {
    hipMemsetAsync(sums, 0, 2 * HH * sizeof(float), stream);
    k_agg<<<dim3(SS/16, BB), 128, 0, stream>>>(adj, deg, cur, agg);
    k_gcn<<<NROWS/16, 128, 0, stream>>>(agg, cur, wl, bl, wr, outb, sums, l);
    k_bn <<<NROWS*HH/256, 256, 0, stream>>>(outb, sums, gmm, bta, wg, nxt, acc, l);
    float* tmp = cur; cur = nxt; nxt = tmp;
  }

  k_head<<<BB, 128, 0, stream>>>(acc, bg, we, be, wo, bo, (float*)d_out);
}